// HET_HGTLayerHetero_33663953666865
// MI455X (gfx1250) — compile-verified
//
#include <hip/hip_runtime.h>
#include <hip/hip_bf16.h>

#define N_NODES 51200
#define N_EDGES 640000
#define N_TYPES 4
#define N_RELS  8
#define N_HEADS 4
#define D_K     32
#define DIM     128
#define NODES_PER_TYPE (N_NODES / N_TYPES)   // 12800
#define EDGES_PER_REL  (N_EDGES / N_RELS)    // 80000
#define RSQRT_DK 0.17677669529663687f        // 1/sqrt(32)

typedef __attribute__((ext_vector_type(2))) float v2f;
typedef __attribute__((ext_vector_type(8))) float v8f;

// D = A(16x4,f32) * B(4x16,f32) + C(16x16,f32)  -> v_wmma_f32_16x16x4_f32
__device__ __forceinline__ v8f wmma_k4(v2f a, v2f b, v8f c) {
  return __builtin_amdgcn_wmma_f32_16x16x4_f32(false, a, false, b, (short)0, c,
                                               false, false);
}

// order-preserving float<->uint encoding for atomicMax-based segment max
__device__ __forceinline__ unsigned enc_f(float f) {
  unsigned u = __float_as_uint(f);
  return (u & 0x80000000u) ? ~u : (u | 0x80000000u);
}
__device__ __forceinline__ float dec_f(unsigned u) {
  return __uint_as_float((u & 0x80000000u) ? (u ^ 0x80000000u) : ~u);
}

// ---------------------------------------------------------------------------
// zero-init a contiguous 32-bit region
__global__ void zero_kernel(unsigned* __restrict__ p, int n) {
  int i = blockIdx.x * blockDim.x + threadIdx.x;
  if (i < n) p[i] = 0u;
}

// ---------------------------------------------------------------------------
// Y[M x 128] = X[M x 128] @ W[type(m)][128 x 128]  (optionally * sigmoid(skip[t]))
// block = 256 threads = 8 waves; wave w -> col tile n0 = 16*w; blockIdx.x -> row tile.
// W[t] (64KB) and the 16x128 X tile (8KB) are staged in LDS; WMMA fragments are
// fed by ds_load, keeping the VMEM path free.
__global__ void typed_gemm128(const float* __restrict__ X,
                              const float* __restrict__ W,
                              const float* __restrict__ skip,   // [T] or nullptr
                              float* __restrict__ Y) {
  __shared__ float sW[DIM * DIM];   // 64 KB
  __shared__ float sX[16 * DIM];    // 8 KB

  const int tid  = threadIdx.x;
  const int m0   = blockIdx.x * 16;
  const int t    = m0 / NODES_PER_TYPE;        // 12800 % 16 == 0: no straddle

  // cooperative coalesced staging (float4 = b128 loads)
  {
    const float4* Wg = (const float4*)(W + (size_t)t * DIM * DIM);
    float4* sW4 = (float4*)sW;
#pragma unroll
    for (int i = 0; i < 16; ++i) sW4[tid + 256 * i] = Wg[tid + 256 * i];
    const float4* Xg = (const float4*)(X + (size_t)m0 * DIM);
    float4* sX4 = (float4*)sX;
#pragma unroll
    for (int i = 0; i < 2; ++i) sX4[tid + 256 * i] = Xg[tid + 256 * i];
  }
  __syncthreads();

  const int lane = tid & 31;
  const int wave = tid >> 5;
  const int half = lane >> 4;
  const int r    = lane & 15;
  const int n0   = wave * 16;

  const float* xrow = sX + r * DIM + 2 * half;
  const float* wcol = sW + (2 * half) * DIM + n0 + r;

  v8f acc = {};
#pragma unroll
  for (int k0 = 0; k0 < DIM; k0 += 4) {
    v2f a, b;
    a.x = xrow[k0];
    a.y = xrow[k0 + 1];
    b.x = wcol[k0 * DIM];
    b.y = wcol[(k0 + 1) * DIM];
    acc = wmma_k4(a, b, acc);
  }

  float g = 1.0f;
  if (skip) g = 1.0f / (1.0f + __expf(-skip[t]));

#pragma unroll
  for (int i = 0; i < 8; ++i) {
    int m = m0 + i + 8 * half;
    Y[(size_t)m * DIM + n0 + r] = g * acc[i];
  }
}

// ---------------------------------------------------------------------------
// Edge attention scores.  Block = 8 waves = 2 edge-tiles x 4 heads, one relation.
// W_att[rel] (all heads, 16KB) staged in LDS.  Per wave:
// aw = q[col] @ W_att[rel,head] (16x32 via two 16x16 WMMA chains, K=32),
// score = sum_d aw*k[row] * pri/sqrt(dk); atomicMax for segment max.
__global__ void hgt_edge_score(const float* __restrict__ qbuf,
                               const float* __restrict__ kbuf,
                               const float* __restrict__ Watt,   // [R,H,32,32]
                               const float* __restrict__ pri,    // [R,H]
                               const int* __restrict__ row_idx,
                               const int* __restrict__ col_idx,
                               float* __restrict__ scores,       // [E,H]
                               unsigned* __restrict__ mmax) {    // [N,H]
  __shared__ float sW[N_HEADS * D_K * D_K];    // 16 KB

  const int tid   = threadIdx.x;
  const int tile0 = blockIdx.x * 2;
  const int rel   = (tile0 * 16) / EDGES_PER_REL;   // 80000 % 32 == 0

  {
    const float4* Wg = (const float4*)(Watt + (size_t)rel * N_HEADS * D_K * D_K);
    float4* sW4 = (float4*)sW;
#pragma unroll
    for (int i = 0; i < 4; ++i) sW4[tid + 256 * i] = Wg[tid + 256 * i];
  }
  __syncthreads();

  const int lane = tid & 31;
  const int wave = tid >> 5;
  const int tile = tile0 + (wave >> 2);
  const int head = wave & 3;
  const int half = lane >> 4;
  const int r    = lane & 15;
  const int e0   = tile * 16;

  const int cA = col_idx[e0 + r];              // dst node feeding A-row r
  const float* qrow = qbuf + (size_t)cA * DIM + head * D_K + 2 * half;
  const float* Wb = sW + head * D_K * D_K + (2 * half) * D_K + r;

  v8f acc0 = {}, acc1 = {};
#pragma unroll
  for (int k0 = 0; k0 < D_K; k0 += 4) {
    v2f a, b0, b1;
    a.x  = qrow[k0];
    a.y  = qrow[k0 + 1];
    b0.x = Wb[k0 * D_K];
    b0.y = Wb[(k0 + 1) * D_K];
    b1.x = Wb[k0 * D_K + 16];
    b1.y = Wb[(k0 + 1) * D_K + 16];
    acc0 = wmma_k4(a, b0, acc0);
    acc1 = wmma_k4(a, b1, acc1);
  }

  const float scale = pri[rel * N_HEADS + head] * RSQRT_DK;
#pragma unroll
  for (int g = 0; g < 8; ++g) {
    const int m  = g + 8 * half;
    const int e  = e0 + m;
    const int rn = row_idx[e];
    float kv0 = kbuf[(size_t)rn * DIM + head * D_K + r];
    float kv1 = kbuf[(size_t)rn * DIM + head * D_K + 16 + r];
    float p = acc0[g] * kv0 + acc1[g] * kv1;
    p += __shfl_xor(p, 1, 32);
    p += __shfl_xor(p, 2, 32);
    p += __shfl_xor(p, 4, 32);
    p += __shfl_xor(p, 8, 32);                 // sum over 16-lane half
    if (r == g) {
      float sc = p * scale;
      scores[(size_t)e * N_HEADS + head] = sc;
      atomicMax(&mmax[(size_t)col_idx[e] * N_HEADS + head], enc_f(sc));
    }
  }
}

// ---------------------------------------------------------------------------
// e = exp(score - max[dst]); scores <- e; ssum[dst] += e
__global__ void hgt_expsum(float* __restrict__ scores,
                           const unsigned* __restrict__ mmax,
                           float* __restrict__ ssum,
                           const int* __restrict__ col_idx) {
  int i = blockIdx.x * blockDim.x + threadIdx.x;       // < E*H
  int e = i >> 2, head = i & 3;
  int cn = col_idx[e];
  float md = dec_f(mmax[(size_t)cn * N_HEADS + head]);
  float ex = __expf(scores[i] - md);
  scores[i] = ex;
  atomicAdd(&ssum[(size_t)cn * N_HEADS + head], ex);
}

// ---------------------------------------------------------------------------
// msg = v[row] @ W_msg[rel,head]; agg[dst] += alpha * msg (atomic scatter).
// Same blocking/LDS staging as hgt_edge_score.
__global__ void hgt_aggregate(const float* __restrict__ vbuf,
                              const float* __restrict__ Wmsg,   // [R,H,32,32]
                              const float* __restrict__ scores, // exp'd
                              const float* __restrict__ ssum,
                              const int* __restrict__ row_idx,
                              const int* __restrict__ col_idx,
                              float* __restrict__ agg) {        // [N,128]
  __shared__ float sW[N_HEADS * D_K * D_K];    // 16 KB

  const int tid   = threadIdx.x;
  const int tile0 = blockIdx.x * 2;
  const int rel   = (tile0 * 16) / EDGES_PER_REL;

  {
    const float4* Wg = (const float4*)(Wmsg + (size_t)rel * N_HEADS * D_K * D_K);
    float4* sW4 = (float4*)sW;
#pragma unroll
    for (int i = 0; i < 4; ++i) sW4[tid + 256 * i] = Wg[tid + 256 * i];
  }
  __syncthreads();

  const int lane = tid & 31;
  const int wave = tid >> 5;
  const int tile = tile0 + (wave >> 2);
  const int head = wave & 3;
  const int half = lane >> 4;
  const int r    = lane & 15;
  const int e0   = tile * 16;

  const int rA = row_idx[e0 + r];              // src node feeding A-row r
  const float* vrow = vbuf + (size_t)rA * DIM + head * D_K + 2 * half;
  const float* Wb = sW + head * D_K * D_K + (2 * half) * D_K + r;

  v8f acc0 = {}, acc1 = {};
#pragma unroll
  for (int k0 = 0; k0 < D_K; k0 += 4) {
    v2f a, b0, b1;
    a.x  = vrow[k0];
    a.y  = vrow[k0 + 1];
    b0.x = Wb[k0 * D_K];
    b0.y = Wb[(k0 + 1) * D_K];
    b1.x = Wb[k0 * D_K + 16];
    b1.y = Wb[(k0 + 1) * D_K + 16];
    acc0 = wmma_k4(a, b0, acc0);
    acc1 = wmma_k4(a, b1, acc1);
  }

#pragma unroll
  for (int g = 0; g < 8; ++g) {
    const int m  = g + 8 * half;
    const int e  = e0 + m;
    const int cn = col_idx[e];
    float al = scores[(size_t)e * N_HEADS + head]
             / (ssum[(size_t)cn * N_HEADS + head] + 1e-16f);
    atomicAdd(&agg[(size_t)cn * DIM + head * D_K + r],      al * acc0[g]);
    atomicAdd(&agg[(size_t)cn * DIM + head * D_K + 16 + r], al * acc1[g]);
  }
}

// ---------------------------------------------------------------------------
extern "C" void kernel_launch(void* const* d_in, const int* in_sizes, int n_in,
                              void* d_out, int out_size, void* d_ws, size_t ws_size,
                              hipStream_t stream) {
  const float* h         = (const float*)d_in[0];
  const float* k_linears = (const float*)d_in[1];
  const float* q_linears = (const float*)d_in[2];
  const float* v_linears = (const float*)d_in[3];
  const float* a_linears = (const float*)d_in[4];
  const float* rel_att   = (const float*)d_in[5];
  const float* rel_msg   = (const float*)d_in[6];
  const float* rel_pri   = (const float*)d_in[7];
  const float* skip      = (const float*)d_in[8];
  const int*   row_idx   = (const int*)d_in[9];
  const int*   col_idx   = (const int*)d_in[10];
  float* out = (float*)d_out;

  // workspace layout (fp32 words)
  float* ws = (float*)d_ws;
  float*    kbuf   = ws;                              // N*DIM
  float*    qbuf   = kbuf + (size_t)N_NODES * DIM;    // N*DIM
  float*    vbuf   = qbuf + (size_t)N_NODES * DIM;    // N*DIM
  float*    scores = vbuf + (size_t)N_NODES * DIM;    // E*H
  unsigned* mmax   = (unsigned*)(scores + (size_t)N_EDGES * N_HEADS); // N*H
  float*    ssum   = (float*)(mmax + (size_t)N_NODES * N_HEADS);      // N*H
  float*    agg    = ssum + (size_t)N_NODES * N_HEADS;                // N*DIM

  // 1) zero mmax / ssum / agg (contiguous)
  {
    int nwords = N_NODES * N_HEADS * 2 + N_NODES * DIM;
    zero_kernel<<<(nwords + 255) / 256, 256, 0, stream>>>(mmax, nwords);
  }

  // 2) K/Q/V typed projections (WMMA f32 GEMM, LDS-staged weights)
  {
    dim3 grid(N_NODES / 16), block(256);
    typed_gemm128<<<grid, block, 0, stream>>>(h, k_linears, nullptr, kbuf);
    typed_gemm128<<<grid, block, 0, stream>>>(h, q_linears, nullptr, qbuf);
    typed_gemm128<<<grid, block, 0, stream>>>(h, v_linears, nullptr, vbuf);
  }

  // 3) edge scores + segment max (2 tiles x 4 heads per block)
  {
    int blocks = (N_EDGES / 16) / 2;                  // 20000
    hgt_edge_score<<<blocks, 256, 0, stream>>>(qbuf, kbuf, rel_att, rel_pri,
                                               row_idx, col_idx, scores, mmax);
  }

  // 4) exp + segment sum
  {
    int n = N_EDGES * N_HEADS;
    hgt_expsum<<<n / 256, 256, 0, stream>>>(scores, mmax, ssum, col_idx);
  }

  // 5) messages + weighted scatter aggregation
  {
    int blocks = (N_EDGES / 16) / 2;
    hgt_aggregate<<<blocks, 256, 0, stream>>>(vbuf, rel_msg, scores, ssum,
                                              row_idx, col_idx, agg);
  }

  // 6) output projection with sigmoid(skip) gate
  {
    dim3 grid(N_NODES / 16), block(256);
    typed_gemm128<<<grid, block, 0, stream>>>(agg, a_linears, skip, out);
  }
}